// VolumetricRenderer_36816459662065
// MI455X (gfx1250) — compile-verified
//
#include <hip/hip_runtime.h>

// Volumetric (NeRF-style) renderer for MI455X / gfx1250.
// Bandwidth-bound stream (~384 MiB @ 23.3 TB/s => ~17us floor).
// Design: 1 wave (32 lanes) per ray, 4 samples/lane.
//   - async global->LDS double-buffered prefetch (ASYNCcnt path)
//   - wave32 exclusive-cumprod scan + reductions via ds_bpermute
//   - v_exp_f32 transcendental
// Outputs (flat, in reference return order):
//   comp_rgb [N,3] | depth [N] | opacity [N] | weights [N,128]

#define N_SAMPLES     128
#define WAVES_PER_BLK 8
#define RAYS_PER_WAVE 8
#define BIG_DELTA     1e10f
#define EPS_T         1e-10f

// Per-ray LDS tile: rgb 384 f32 | sigma 128 f32 | z 128 f32  = 640 f32 = 2560 B
#define BUF_FLOATS    640
#define RGB_OFF_B     0u
#define SIG_OFF_B     1536u
#define Z_OFF_B       2048u

__device__ __forceinline__ float bperm(float x, int srcLane) {
  // wave32: ds_bpermute uses index bits [6:2]; callers guard out-of-range.
  return __int_as_float(
      __builtin_amdgcn_ds_bpermute(srcLane << 2, __float_as_int(x)));
}

__device__ __forceinline__ void async_ld16(unsigned ldsByte, const void* g) {
  // CDNA5 async copy: GLOBAL_LOAD_ASYNC_TO_LDS_B128 (tracked by ASYNCcnt).
  // vdst = LDS byte address (wave-relative), vaddr = 64-bit global address.
  asm volatile("global_load_async_to_lds_b128 %0, %1, off"
               :: "v"(ldsByte), "v"(g)
               : "memory");
}

template <int N>
__device__ __forceinline__ void wait_async() {
#if __has_builtin(__builtin_amdgcn_s_wait_asynccnt)
  __builtin_amdgcn_s_wait_asynccnt(N);
#else
  asm volatile("s_wait_asynccnt %0" :: "i"(N) : "memory");
#endif
  asm volatile("" ::: "memory");   // keep LDS reads below the wait
}

// Issue the 5 async-load instructions that stage one ray's data into LDS.
__device__ __forceinline__ void prefetch_ray(unsigned bufByte,
                                             const float* rgb,
                                             const float* sig,
                                             const float* zv,
                                             int lane) {
  // rgb: 1536 B = 96 x b128; lane handles chunks lane, lane+32, lane+64
#pragma unroll
  for (int k = 0; k < 3; ++k)
    async_ld16(bufByte + RGB_OFF_B + 16u * (unsigned)(lane + 32 * k),
               rgb + 4 * lane + 128 * k);
  async_ld16(bufByte + SIG_OFF_B + 16u * (unsigned)lane, sig + 4 * lane);
  async_ld16(bufByte + Z_OFF_B   + 16u * (unsigned)lane, zv  + 4 * lane);
}

__global__ __launch_bounds__(WAVES_PER_BLK * 32)
void nerf_render_kernel(const float* __restrict__ rgb,
                        const float* __restrict__ sigma,
                        const float* __restrict__ zvals,
                        const int*   __restrict__ whitep,
                        float* __restrict__ out,
                        int nray) {
  __shared__ __align__(16) float smem[WAVES_PER_BLK * 2 * BUF_FLOATS];

  const int lane = (int)(threadIdx.x & 31u);
  const int wave = (int)(threadIdx.x >> 5);
  const int wgid = (int)blockIdx.x * WAVES_PER_BLK + wave;   // global wave id
  const int ray0 = wgid * RAYS_PER_WAVE;
  const int white = *whitep;

  float* __restrict__ out_rgb   = out;                       // [N,3]
  float* __restrict__ out_depth = out + (size_t)nray * 3;    // [N]
  float* __restrict__ out_opac  = out + (size_t)nray * 4;    // [N]
  float* __restrict__ out_wts   = out + (size_t)nray * 5;    // [N,128]

  const unsigned smemBase = (unsigned)(size_t)(&smem[0]);    // LDS byte offset
  const unsigned waveBase = smemBase + (unsigned)(wave * 2 * BUF_FLOATS) * 4u;
  const float*   waveF    = smem + wave * 2 * BUF_FLOATS;

  // ---- prologue prefetch: ray0 into buffer 0 ----
  if (ray0 < nray)
    prefetch_ray(waveBase, rgb + (size_t)ray0 * 384,
                 sigma + (size_t)ray0 * 128, zvals + (size_t)ray0 * 128, lane);

#pragma unroll 1
  for (int i = 0; i < RAYS_PER_WAVE; ++i) {
    const int ray = ray0 + i;
    const int b   = i & 1;

    const bool haveNext = (i + 1 < RAYS_PER_WAVE) && (ray0 + i + 1 < nray);
    if (haveNext) {
      const int rn = ray + 1;
      prefetch_ray(waveBase + (unsigned)((1 - b) * BUF_FLOATS) * 4u,
                   rgb + (size_t)rn * 384, sigma + (size_t)rn * 128,
                   zvals + (size_t)rn * 128, lane);
      wait_async<5>();   // 5 async ops outstanding for ray i+1; ray i complete
    } else {
      wait_async<0>();
    }

    if (ray >= nray) continue;

    // ---- read this ray's tile from LDS (ds_load_b128) ----
    const float4* bufv =
        reinterpret_cast<const float4*>(waveF + b * BUF_FLOATS);
    const float4 c0 = bufv[3 * lane + 0];   // {r0 g0 b0 r1}
    const float4 c1 = bufv[3 * lane + 1];   // {g1 b1 r2 g2}
    const float4 c2 = bufv[3 * lane + 2];   // {b2 r3 g3 b3}
    const float4 s4 = bufv[96  + lane];     // sigma[4l..4l+3]
    const float4 z4 = bufv[128 + lane];     // z[4l..4l+3]

    // ---- deltas (neighbor z via cross-lane) ----
    const float znext = bperm(z4.x, lane + 1);     // lane31 value unused
    const float d0 = z4.y - z4.x;
    const float d1 = z4.z - z4.y;
    const float d2 = z4.w - z4.z;
    const float d3 = (lane == 31) ? BIG_DELTA : (znext - z4.w);

    // ---- transmittance terms: e = exp(-delta*relu(sigma)) = 1 - alpha ----
    const float e0 = __expf(-d0 * fmaxf(s4.x, 0.0f));
    const float e1 = __expf(-d1 * fmaxf(s4.y, 0.0f));
    const float e2 = __expf(-d2 * fmaxf(s4.z, 0.0f));
    const float e3 = __expf(-d3 * fmaxf(s4.w, 0.0f));
    const float t0 = e0 + EPS_T, t1 = e1 + EPS_T;
    const float t2 = e2 + EPS_T, t3 = e3 + EPS_T;

    // ---- exclusive cumulative product over 128 samples ----
    // local exclusive prefixes within the lane's 4 samples
    const float p1 = t0, p2 = t0 * t1, p3 = p2 * t2;
    float pref = p3 * t3;                       // lane-local total
#pragma unroll
    for (int d = 1; d < 32; d <<= 1) {          // inclusive wave scan
      const float o = bperm(pref, lane - d);
      pref = (lane >= d) ? pref * o : pref;
    }
    float excl = bperm(pref, lane - 1);         // exclusive across lanes
    excl = (lane == 0) ? 1.0f : excl;

    // ---- weights ----
    const float w0 = (1.0f - e0) * excl;
    const float w1 = (1.0f - e1) * excl * p1;
    const float w2 = (1.0f - e2) * excl * p2;
    const float w3 = (1.0f - e3) * excl * p3;

    *reinterpret_cast<float4*>(out_wts + (size_t)ray * N_SAMPLES + 4 * lane) =
        make_float4(w0, w1, w2, w3);

    // ---- per-lane partial sums ----
    float pr = w0 * c0.x + w1 * c0.w + w2 * c1.z + w3 * c2.y;
    float pg = w0 * c0.y + w1 * c1.x + w2 * c1.w + w3 * c2.z;
    float pb = w0 * c0.z + w1 * c1.y + w2 * c2.x + w3 * c2.w;
    float pd = w0 * z4.x + w1 * z4.y + w2 * z4.z + w3 * z4.w;
    float po = w0 + w1 + w2 + w3;

    // ---- wave32 butterfly reduction ----
#pragma unroll
    for (int m = 16; m >= 1; m >>= 1) {
      pr += bperm(pr, lane ^ m);
      pg += bperm(pg, lane ^ m);
      pb += bperm(pb, lane ^ m);
      pd += bperm(pd, lane ^ m);
      po += bperm(po, lane ^ m);
    }

    if (lane == 0) {
      const float bg = white ? (1.0f - po) : 0.0f;
      out_rgb[(size_t)ray * 3 + 0] = pr + bg;
      out_rgb[(size_t)ray * 3 + 1] = pg + bg;
      out_rgb[(size_t)ray * 3 + 2] = pb + bg;
      out_depth[ray] = pd;
      out_opac[ray]  = po;
    }
  }
}

extern "C" void kernel_launch(void* const* d_in, const int* in_sizes, int n_in,
                              void* d_out, int out_size, void* d_ws,
                              size_t ws_size, hipStream_t stream) {
  const float* rgb   = (const float*)d_in[0];   // [N,128,3]
  const float* sigma = (const float*)d_in[1];   // [N,128]
  const float* zv    = (const float*)d_in[2];   // [N,128]
  const int*   white = (const int*)d_in[3];     // scalar
  float* out = (float*)d_out;

  const int nray = in_sizes[0] / (N_SAMPLES * 3);
  const int raysPerBlock = WAVES_PER_BLK * RAYS_PER_WAVE;    // 64
  const int blocks = (nray + raysPerBlock - 1) / raysPerBlock;

  nerf_render_kernel<<<blocks, WAVES_PER_BLK * 32, 0, stream>>>(
      rgb, sigma, zv, white, out, nray);
}